// MoM_64991445123771
// MI455X (gfx1250) — compile-verified
//
#include <hip/hip_runtime.h>
#include <hip/hip_bf16.h>

typedef __attribute__((ext_vector_type(2))) float v2f;
typedef __attribute__((ext_vector_type(4))) float v4f;
typedef __attribute__((ext_vector_type(8))) float v8f;

#define HEADS 12
#define EXPERTS 3
#define HD 64
#define DIN 768          // HEADS * HD
#define DOUT 2304        // EXPERTS * HEADS * HD
#define PITCH 68         // LDS row pitch in floats (64 + 4): bank-conflict-free b64 reads, keeps 16B alignment
#define ROWS_PER_BLOCK 128

__global__ __launch_bounds__(256) void mom_wmma_f32_kernel(
    const float* __restrict__ x1, const float* __restrict__ x2, const float* __restrict__ x3,
    const float* __restrict__ W,  const float* __restrict__ bias,
    const float* __restrict__ gamma, const float* __restrict__ beta,
    const float* __restrict__ rmean, const float* __restrict__ rvar,
    float* __restrict__ out)
{
    __shared__ float Wl[HD * PITCH];   // 64x64 weights, padded pitch (~17 KB)
    __shared__ float biasL[HD];
    __shared__ float sL[HD];           // gamma * rsqrt(var + eps)
    __shared__ float tL[HD];           // beta - mean * s

    const int he = blockIdx.y;
    const int h  = he / EXPERTS;
    const int e  = he - h * EXPERTS;
    const float* __restrict__ X = (e == 0) ? x1 : ((e == 1) ? x2 : x3);

    const int tid = threadIdx.x;

    // ---- Stage W[h][e] (64x64 f32) into LDS: 1024 float4s over 256 threads ----
    const float* __restrict__ Wg = W + (size_t)he * (HD * HD);
    #pragma unroll
    for (int j = 0; j < 4; ++j) {
        const int e4 = tid + j * 256;      // float4 index 0..1023
        const int f  = e4 >> 4;            // output-feature row 0..63
        const int d4 = (e4 & 15) << 2;     // input-dim col 0,4,...,60
        v4f v = *(const v4f*)(Wg + f * HD + d4);
        *(v4f*)(&Wl[f * PITCH + d4]) = v;
    }
    // ---- Fused epilogue params ----
    if (tid < HD) {
        const int p = he * HD + tid;
        const float s = gamma[p] * rsqrtf(rvar[p] + 1e-5f);
        sL[tid]    = s;
        tL[tid]    = beta[p] - rmean[p] * s;
        biasL[tid] = bias[p];
    }
    __syncthreads();

    // ---- Per-wave 16x64 output tile ----
    const int lane = tid & 31;
    const int wave = tid >> 5;
    const int n15  = lane & 15;
    const int half = lane >> 4;        // 0: lanes 0-15, 1: lanes 16-31
    const int koff = half << 1;        // A/B fragment K offset: 0 or 2

    const int rowBase = blockIdx.x * ROWS_PER_BLOCK + wave * 16;
    const float* __restrict__ Xrow = X + (size_t)(rowBase + n15) * DIN + h * HD;

    v8f acc[4];
    acc[0] = (v8f)(0.0f); acc[1] = (v8f)(0.0f); acc[2] = (v8f)(0.0f); acc[3] = (v8f)(0.0f);

    #pragma unroll
    for (int kb = 0; kb < 16; ++kb) {
        // A fragment: row = lane&15, K = kb*4 + {koff, koff+1}  (8B contiguous)
        v2f a = *(const v2f*)(Xrow + kb * 4 + koff);
        const float* wb = &Wl[kb * 4 + koff];
        // B fragments: B[k][n] = W[n][k]; contiguous along d(=K) in LDS
        v2f b0 = *(const v2f*)(wb + (0 * 16 + n15) * PITCH);
        v2f b1 = *(const v2f*)(wb + (1 * 16 + n15) * PITCH);
        v2f b2 = *(const v2f*)(wb + (2 * 16 + n15) * PITCH);
        v2f b3 = *(const v2f*)(wb + (3 * 16 + n15) * PITCH);
        acc[0] = __builtin_amdgcn_wmma_f32_16x16x4_f32(false, a, false, b0, (short)0, acc[0], false, false);
        acc[1] = __builtin_amdgcn_wmma_f32_16x16x4_f32(false, a, false, b1, (short)0, acc[1], false, false);
        acc[2] = __builtin_amdgcn_wmma_f32_16x16x4_f32(false, a, false, b2, (short)0, acc[2], false, false);
        acc[3] = __builtin_amdgcn_wmma_f32_16x16x4_f32(false, a, false, b3, (short)0, acc[3], false, false);
    }

    // ---- Fused bias + QuickGELU + BatchNorm(eval) + strided store ----
    // C/D layout: VGPR r holds M = r (lanes 0-15) / M = r+8 (lanes 16-31), N = lane&15
    const size_t colBase = (size_t)e * (HEADS * HD) + (size_t)h * HD;
    const int mhalf = half << 3;
    #pragma unroll
    for (int nt = 0; nt < 4; ++nt) {
        const int n = nt * 16 + n15;
        const float bs = biasL[n];
        const float sc = sL[n];
        const float sh = tL[n];
        float* __restrict__ obase = out + colBase + n;
        #pragma unroll
        for (int r = 0; r < 8; ++r) {
            float v = acc[nt][r] + bs;
            float g = v * (1.0f / (1.0f + __expf(-1.702f * v)));   // QuickGELU
            float o = g * sc + sh;                                  // BN eval (fused affine)
            obase[(size_t)(rowBase + mhalf + r) * DOUT] = o;
        }
    }
}

extern "C" void kernel_launch(void* const* d_in, const int* in_sizes, int n_in,
                              void* d_out, int out_size, void* d_ws, size_t ws_size,
                              hipStream_t stream) {
    const float* x1    = (const float*)d_in[0];
    const float* x2    = (const float*)d_in[1];
    const float* x3    = (const float*)d_in[2];
    const float* W     = (const float*)d_in[3];
    const float* bias  = (const float*)d_in[4];
    const float* gamma = (const float*)d_in[5];
    const float* beta  = (const float*)d_in[6];
    const float* rmean = (const float*)d_in[7];
    const float* rvar  = (const float*)d_in[8];
    float* out = (float*)d_out;

    const int B = in_sizes[0] / DIN;                 // 65536
    dim3 grid(B / ROWS_PER_BLOCK, HEADS * EXPERTS);  // (512, 36)
    mom_wmma_f32_kernel<<<grid, 256, 0, stream>>>(
        x1, x2, x3, W, bias, gamma, beta, rmean, rvar, out);
}